// ProbAttention_42356967473350
// MI455X (gfx1250) — compile-verified
//
#include <hip/hip_runtime.h>
#include <math.h>

#define B_    4
#define H_    8
#define L_    2048
#define D_    64
#define SK_   40
#define NTOP_ 40
#define NSPLIT_ 8            // key-range splits for the attention kernel
#define KPS_  (L_ / NSPLIT_) // 256 keys per split
#define NCHUNK_ 16           // cumsum chunks
#define CROWS_ (L_ / NCHUNK_)
#define SCALE_ 0.125f        // 1/sqrt(64)

typedef float v2f __attribute__((ext_vector_type(2)));
typedef float v8f __attribute__((ext_vector_type(8)));

// ---------------------------------------------------------------------------
// Kernel 1: sparsity metric M[b,h,l] = max_s(q.k_s) - sum_s(q.k_s)/L
// ---------------------------------------------------------------------------
__global__ void compute_m_kernel(const float* __restrict__ q,
                                 const float* __restrict__ k,
                                 const int*   __restrict__ idx,
                                 float*       __restrict__ m_out) {
  int g  = blockIdx.x * blockDim.x + threadIdx.x;   // g = bh*L + l
  int l  = g & (L_ - 1);
  int bh = g >> 11;
  const float* qp = q + (size_t)g * D_;
  float qr[D_];
#pragma unroll
  for (int d = 0; d < D_; d += 4) {
    float4 t = *(const float4*)(qp + d);
    qr[d] = t.x; qr[d + 1] = t.y; qr[d + 2] = t.z; qr[d + 3] = t.w;
  }
  const float* kbase = k + (size_t)bh * L_ * D_;
  float mx = -INFINITY, sm = 0.f;
  for (int s = 0; s < SK_; ++s) {
    int ki = idx[l * SK_ + s];
    const float* kp = kbase + (size_t)ki * D_;
    float acc = 0.f;
#pragma unroll
    for (int d = 0; d < D_; d += 4) {
      float4 t = *(const float4*)(kp + d);
      acc += qr[d] * t.x + qr[d + 1] * t.y + qr[d + 2] * t.z + qr[d + 3] * t.w;
    }
    mx = fmaxf(mx, acc);
    sm += acc;
  }
  m_out[g] = mx - sm * (1.0f / (float)L_);
}

// ---------------------------------------------------------------------------
// Kernel 2: top-40 of M per (b,h) via iterative argmax (ties -> smaller idx).
// ---------------------------------------------------------------------------
__global__ void topk_kernel(const float* __restrict__ m_in,
                            int*         __restrict__ top_out) {
  __shared__ float vals[L_];
  __shared__ float rv[256];
  __shared__ int   ri[256];
  int bh  = blockIdx.x;
  int tid = threadIdx.x;
  for (int i = tid; i < L_; i += 256) vals[i] = m_in[(size_t)bh * L_ + i];
  __syncthreads();
  for (int sel = 0; sel < NTOP_; ++sel) {
    float bv = -INFINITY; int bi = 0x7FFFFFFF;
    for (int i = tid; i < L_; i += 256) {
      float vv = vals[i];
      if (vv > bv) { bv = vv; bi = i; }
    }
    rv[tid] = bv; ri[tid] = bi;
    __syncthreads();
    for (int off = 128; off > 0; off >>= 1) {
      if (tid < off) {
        float ov = rv[tid + off]; int oi = ri[tid + off];
        if (ov > rv[tid] || (ov == rv[tid] && oi < ri[tid])) {
          rv[tid] = ov; ri[tid] = oi;
        }
      }
      __syncthreads();
    }
    if (tid == 0) {
      top_out[bh * NTOP_ + sel] = ri[0];
      vals[ri[0]] = -INFINITY;
    }
    __syncthreads();
  }
}

// ---------------------------------------------------------------------------
// Kernel 3a/3b: two-pass chunked cumsum(values, axis=2) -> 512 WGs.
// ---------------------------------------------------------------------------
__global__ void cumsum_part_kernel(const float* __restrict__ v,
                                   float* __restrict__ csum) {
  int bh = blockIdx.x, ch = blockIdx.y, d = threadIdx.x;
  const float* vp = v + ((size_t)bh * L_ + (size_t)ch * CROWS_) * D_ + d;
  float s = 0.f;
  for (int l = 0; l < CROWS_; ++l) s += vp[(size_t)l * D_];
  csum[(bh * NCHUNK_ + ch) * D_ + d] = s;
}

__global__ void cumsum_scan_kernel(const float* __restrict__ v,
                                   const float* __restrict__ csum,
                                   float* __restrict__ out) {
  int bh = blockIdx.x, ch = blockIdx.y, d = threadIdx.x;
  float acc = 0.f;
  for (int c = 0; c < ch; ++c) acc += csum[(bh * NCHUNK_ + c) * D_ + d];
  const float* vp = v   + ((size_t)bh * L_ + (size_t)ch * CROWS_) * D_ + d;
  float*       op = out + ((size_t)bh * L_ + (size_t)ch * CROWS_) * D_ + d;
  for (int l = 0; l < CROWS_; ++l) {
    acc += vp[(size_t)l * D_];
    op[(size_t)l * D_] = acc;
  }
}

// ---------------------------------------------------------------------------
// Kernel 4: split-K flash attention partials. grid (bh=32, split=8), 96 thr.
// K/V tiles staged to LDS via async global->LDS copies (ASYNCcnt), double
// buffered so copies of tile t+1 overlap WMMAs of tile t. All matrix math
// via V_WMMA_F32_16X16X4_F32.
// ---------------------------------------------------------------------------
__device__ __forceinline__ void async_tile_copy(const float* gk, const float* gv,
                                                float* lk, float* lv, int tid) {
  // copy two 16x64 f32 tiles (4KB each) with per-lane B128 async transfers
  for (int i = tid; i < 256; i += 96) {
    unsigned lka = (unsigned)(size_t)(lk + i * 4);
    unsigned lva = (unsigned)(size_t)(lv + i * 4);
    unsigned long long gka = (unsigned long long)(size_t)(gk + i * 4);
    unsigned long long gva = (unsigned long long)(size_t)(gv + i * 4);
    asm volatile("global_load_async_to_lds_b128 %0, %1, off"
                 :: "v"(lka), "v"(gka) : "memory");
    asm volatile("global_load_async_to_lds_b128 %0, %1, off"
                 :: "v"(lva), "v"(gva) : "memory");
  }
}

__device__ __forceinline__ void wait_async0() {
  asm volatile("s_wait_asynccnt 0" ::: "memory");
}

__global__ __launch_bounds__(96) void attn_split_kernel(
    const float* __restrict__ q, const float* __restrict__ k,
    const float* __restrict__ v, const int* __restrict__ mtop,
    float* __restrict__ part_m, float* __restrict__ part_l,
    float* __restrict__ part_acc) {
  __shared__ int   s_idx[48];
  __shared__ float s_p[3][16 * 16];
  __shared__ __align__(16) float s_k[2][16 * D_];
  __shared__ __align__(16) float s_v[2][16 * D_];

  int bh    = blockIdx.x;
  int split = blockIdx.y;
  int tid   = threadIdx.x;
  int wave  = tid >> 5;
  int lane  = tid & 31;
  int half  = lane >> 4;
  int col   = lane & 15;

  if (tid < 48) s_idx[tid] = (tid < NTOP_) ? mtop[bh * NTOP_ + tid] : 0x7FFFFFFF;
  __syncthreads();

  int u0 = wave * 16;

  // A fragments of Q_reduce: lane t holds row t%16, K-pair (4kk+2*half, +1)
  int ua   = u0 + col;
  int qrow = (ua < NTOP_) ? s_idx[ua] : 0;
  const float* qp = q + ((size_t)bh * L_ + qrow) * D_ + 2 * half;
  float a0[16], a1[16];
#pragma unroll
  for (int kk = 0; kk < 16; ++kk) {
    v2f t = *(const v2f*)(qp + 4 * kk);
    a0[kk] = t.x; a1[kk] = t.y;
  }

  int mrow[8];
#pragma unroll
  for (int r = 0; r < 8; ++r) mrow[r] = s_idx[u0 + r + half * 8];

  float m_run[8], l_run[8];
#pragma unroll
  for (int r = 0; r < 8; ++r) { m_run[r] = -INFINITY; l_run[r] = 0.f; }
  v8f c_out[4] = {};

  int n_base = split * KPS_;
  const float* kbase = k + (size_t)bh * L_ * D_;
  const float* vbase = v + (size_t)bh * L_ * D_;

  // preload tile 0 of this split
  async_tile_copy(kbase + (size_t)n_base * D_, vbase + (size_t)n_base * D_,
                  s_k[0], s_v[0], tid);
  wait_async0();
  __syncthreads();

  const int NT = KPS_ / 16;   // 16 tiles per split
  for (int t = 0; t < NT; ++t) {
    int cur = t & 1;
    if (t + 1 < NT) {         // overlap next tile's copy with this tile's math
      int nn = n_base + (t + 1) * 16;
      async_tile_copy(kbase + (size_t)nn * D_, vbase + (size_t)nn * D_,
                      s_k[1 - cur], s_v[1 - cur], tid);
    }

    // ---- scores tile from LDS-staged K: 16 k-steps of K=4
    v8f c = {};
#pragma unroll
    for (int kk = 0; kk < 16; ++kk) {
      v2f bvec = *(const v2f*)&s_k[cur][col * D_ + 4 * kk + 2 * half];
      v2f avec; avec.x = a0[kk]; avec.y = a1[kk];
      c = __builtin_amdgcn_wmma_f32_16x16x4_f32(false, avec, false, bvec,
                                                (short)0, c, false, false);
    }

    // ---- mask + online softmax
    int kn = n_base + t * 16 + col;
#pragma unroll
    for (int r = 0; r < 8; ++r) {
      float x = (kn > mrow[r]) ? -INFINITY : c[r] * SCALE_;
      float bm = x;
#pragma unroll
      for (int off = 1; off < 16; off <<= 1)
        bm = fmaxf(bm, __shfl_xor(bm, off, 16));
      float mn   = fmaxf(m_run[r], bm);
      float sfac = __expf(m_run[r] - mn);
      float p    = __expf(x - mn);
      float rs = p;
#pragma unroll
      for (int off = 1; off < 16; off <<= 1)
        rs += __shfl_xor(rs, off, 16);
      l_run[r] = l_run[r] * sfac + rs;
      m_run[r] = mn;
#pragma unroll
      for (int j = 0; j < 4; ++j) c_out[j][r] *= sfac;
      s_p[wave][(r + half * 8) * 16 + col] = p;
    }
    __syncthreads();

    // ---- P(16x16) x V(16x64) from LDS-staged V
#pragma unroll
    for (int kk2 = 0; kk2 < 4; ++kk2) {
      int kb = 4 * kk2 + 2 * half;
      v2f pa;
      pa.x = s_p[wave][col * 16 + kb];
      pa.y = s_p[wave][col * 16 + kb + 1];
#pragma unroll
      for (int j = 0; j < 4; ++j) {
        v2f bvec;
        bvec.x = s_v[cur][kb * D_ + 16 * j + col];
        bvec.y = s_v[cur][(kb + 1) * D_ + 16 * j + col];
        c_out[j] = __builtin_amdgcn_wmma_f32_16x16x4_f32(
            false, pa, false, bvec, (short)0, c_out[j], false, false);
      }
    }
    wait_async0();      // next tile's async copies have landed
    __syncthreads();    // everyone done reading cur + s_p before reuse
  }

  // ---- write split partials (m, l, unnormalized acc)
#pragma unroll
  for (int r = 0; r < 8; ++r) {
    int row  = u0 + r + half * 8;                 // 0..47
    int base = (bh * NSPLIT_ + split) * 48 + row;
    if (col == 0) { part_m[base] = m_run[r]; part_l[base] = l_run[r]; }
#pragma unroll
    for (int j = 0; j < 4; ++j)
      part_acc[(size_t)base * D_ + 16 * j + col] = c_out[j][r];
  }
}

// ---------------------------------------------------------------------------
// Kernel 5: merge split partials (flash combine) and scatter into context.
// ---------------------------------------------------------------------------
__global__ void combine_kernel(const int* __restrict__ mtop,
                               const float* __restrict__ part_m,
                               const float* __restrict__ part_l,
                               const float* __restrict__ part_acc,
                               float* __restrict__ out) {
  int bh = blockIdx.x;
  for (int e = threadIdx.x; e < NTOP_ * D_; e += blockDim.x) {
    int u = e >> 6, d = e & 63;
    float mt = -INFINITY;
#pragma unroll
    for (int s = 0; s < NSPLIT_; ++s)
      mt = fmaxf(mt, part_m[(bh * NSPLIT_ + s) * 48 + u]);
    float lt = 0.f, av = 0.f;
#pragma unroll
    for (int s = 0; s < NSPLIT_; ++s) {
      int base = (bh * NSPLIT_ + s) * 48 + u;
      float w = __expf(part_m[base] - mt);
      lt += part_l[base] * w;
      av += part_acc[(size_t)base * D_ + d] * w;
    }
    int row = mtop[bh * NTOP_ + u];
    out[((size_t)bh * L_ + row) * D_ + d] = av / lt;
  }
}

// ---------------------------------------------------------------------------
extern "C" void kernel_launch(void* const* d_in, const int* in_sizes, int n_in,
                              void* d_out, int out_size, void* d_ws, size_t ws_size,
                              hipStream_t stream) {
  (void)in_sizes; (void)n_in; (void)out_size; (void)ws_size;
  const float* q    = (const float*)d_in[0];
  const float* k    = (const float*)d_in[1];
  const float* v    = (const float*)d_in[2];
  const int*   idxs = (const int*)d_in[3];
  float* out = (float*)d_out;

  // workspace layout (floats)
  float* ws       = (float*)d_ws;
  float* m_ws     = ws;                                  // 65536
  float* csum_ws  = ws + 65536;                          // 32768
  int*   top_ws   = (int*)(ws + 98304);                  // 1280 ints
  float* part_m   = ws + 98304 + 1280;                   // 32*8*48 = 12288
  float* part_l   = part_m + 12288;                      // 12288
  float* part_acc = part_l + 12288;                      // 32*8*48*64 = 786432

  compute_m_kernel<<<(B_ * H_ * L_) / 256, 256, 0, stream>>>(q, k, idxs, m_ws);
  topk_kernel<<<B_ * H_, 256, 0, stream>>>(m_ws, top_ws);
  cumsum_part_kernel<<<dim3(B_ * H_, NCHUNK_), D_, 0, stream>>>(v, csum_ws);
  cumsum_scan_kernel<<<dim3(B_ * H_, NCHUNK_), D_, 0, stream>>>(v, csum_ws, out);
  attn_split_kernel<<<dim3(B_ * H_, NSPLIT_), 96, 0, stream>>>(
      q, k, v, top_ws, part_m, part_l, part_acc);
  combine_kernel<<<B_ * H_, 256, 0, stream>>>(top_ws, part_m, part_l, part_acc, out);
}